// StateAttention_2276332667138
// MI455X (gfx1250) — compile-verified
//
#include <hip/hip_runtime.h>
#include <hip/hip_bf16.h>
#include <stdint.h>

// Problem constants (from reference)
#define DFEAT 256   // IN_FEATURES == OUT_FEATURES == INTERNAL == 256
#define NB    32    // batch
#define TH    256   // history length
#define TQ    64    // query length

typedef float v2f __attribute__((ext_vector_type(2)));
typedef float v8f __attribute__((ext_vector_type(8)));

__device__ __forceinline__ v8f wmma4(v2f a, v2f b, v8f c) {
  // D(16x16,f32) = A(16x4,f32) * B(4x16,f32) + C   -- exact f32 matrix pipe
  return __builtin_amdgcn_wmma_f32_16x16x4_f32(false, a, false, b, (short)0, c,
                                               false, false);
}

__device__ __forceinline__ float fast_exp2(float x) {
#if __has_builtin(__builtin_amdgcn_exp2f)
  return __builtin_amdgcn_exp2f(x);
#else
  return exp2f(x);
#endif
}
__device__ __forceinline__ float fast_rcp(float x) {
#if __has_builtin(__builtin_amdgcn_rcpf)
  return __builtin_amdgcn_rcpf(x);
#else
  return 1.0f / x;
#endif
}
__device__ __forceinline__ float fast_tanh(float x) {
#if __has_builtin(__builtin_amdgcn_tanhf)
  return __builtin_amdgcn_tanhf(x);
#else
  // tanh(x) = 1 - 2/(1+e^{2x}); e^{2x} = 2^{x * 2*log2(e)}
  return 1.0f - 2.0f * fast_rcp(1.0f + fast_exp2(x * 2.8853900817779268f));
#endif
}

// CDNA5 async global->LDS copy (ASYNCcnt-tracked, writes LDS directly).
// LDS offset = low 32 bits of the generic pointer (LDS aperture is in the
// high 32 bits per the flat-addressing aperture rules).
__device__ __forceinline__ void async_load_b128(void* lds_ptr, const void* gptr) {
  uint32_t lds_off = (uint32_t)(uintptr_t)lds_ptr;
  unsigned long long ga = (unsigned long long)(uintptr_t)gptr;
  asm volatile("global_load_async_to_lds_b128 %0, %1, off"
               :: "v"(lds_off), "v"(ga)
               : "memory");
}
__device__ __forceinline__ void wait_asynccnt(int n) {
  if (n == 0)
    asm volatile("s_wait_asynccnt 0" ::: "memory");
  else
    asm volatile("s_wait_asynccnt 4" ::: "memory");
}

// ---------------------------------------------------------------------------
// Kernel 1: k_proj[m,d] = sum_i keys[m,i] * w1[d,i]   (M=8192, N=256, K=256)
// One 16x16 output tile per wave; 8 waves per block; 1024 blocks.
// ---------------------------------------------------------------------------
__global__ __launch_bounds__(256) void kproj_gemm(const float* __restrict__ X,
                                                  const float* __restrict__ W,
                                                  float* __restrict__ P) {
  const int tid  = threadIdx.x;
  const int lane = tid & 31;
  const int wave = tid >> 5;
  const int tile = blockIdx.x * 8 + wave;          // 8192 tiles total
  const int nT = (tile & 15) * 16;                 // output col tile (d)
  const int mT = (tile >> 4) * 16;                 // output row tile (m)

  const int r16  = lane & 15;
  const int klo  = (lane < 16) ? 0 : 2;            // A/B K sub-offset per lane half
  const float* arow = X + (size_t)(mT + r16) * DFEAT;      // A[m][k]
  const float* brow = W + (size_t)(nT + r16) * DFEAT;      // B[k][n] = W[n][k]

  v8f c = {};
#pragma unroll
  for (int kk = 0; kk < DFEAT; kk += 4) {
    v2f a = *(const v2f*)(arow + kk + klo);        // contiguous K,K+1 -> b64 load
    v2f b = *(const v2f*)(brow + kk + klo);
    c = wmma4(a, b, c);
  }

  const int rbase = (lane < 16) ? 0 : 8;           // C/D: vgpr r -> rows r, r+8
  const int col = nT + r16;
#pragma unroll
  for (int r = 0; r < 8; ++r)
    P[(size_t)(mT + rbase + r) * DFEAT + col] = c[r];
}

// ---------------------------------------------------------------------------
// Kernel 2: fused q_proj (WMMA) + additive scores (tanh, double-buffered
//           async global->LDS) + softmax + context GEMM (WMMA).
// One block per (batch, 16-query tile).
// ---------------------------------------------------------------------------
#define QP_S 260   // padded LDS row stride (floats): 4*t distinct banks mod 64
#define SC_S 258

__global__ __launch_bounds__(256) void fused_attn(const float* __restrict__ keys,
                                                  const float* __restrict__ queries,
                                                  const float* __restrict__ w2,
                                                  const float* __restrict__ vvec,
                                                  const float* __restrict__ kproj,
                                                  float* __restrict__ out) {
  __shared__ float v_lds[DFEAT];
  __shared__ float qp_lds[16 * QP_S];      // q_proj tile   (16 x 256, padded)
  __shared__ float kp_lds[2][16 * QP_S];   // k_proj chunks (double-buffered)
  __shared__ float sc_lds[16 * SC_S];      // scores -> alpha (16 x 256, padded)

  const int tid  = threadIdx.x;
  const int lane = tid & 31;
  const int wave = tid >> 5;
  const int b     = blockIdx.x >> 2;   // 32 batches
  const int qtile = blockIdx.x & 3;    // 4 query tiles of 16

  const int r16  = lane & 15;
  const int klo  = (lane < 16) ? 0 : 2;
  const int rbase = (lane < 16) ? 0 : 8;

  // --- Phase A: v into LDS; kick off async load of k_proj chunk 0 ---------
  v_lds[tid] = vvec[tid];

  auto issue_chunk = [&](int tc, int buf) {
    const float* kpg = kproj + ((size_t)b * TH + tc * 16) * DFEAT;
#pragma unroll
    for (int it = 0; it < 4; ++it) {             // 4096 floats, b128 per thread
      int flat = (tid + it * 256) * 4;
      int rr = flat >> 8;
      int cc = flat & 255;
      async_load_b128(&kp_lds[buf][rr * QP_S + cc], kpg + flat);
    }
  };
  issue_chunk(0, 0);

  // --- Phase B: q_proj tile via WMMA, straight into LDS -------------------
  // qp[q][d] = sum_o queries[b, qtile*16+q, o] * w2[d, o]
  {
    const float* Qbase = queries + (size_t)(b * TQ + qtile * 16) * DFEAT;
    const float* arow = Qbase + (size_t)r16 * DFEAT;
#pragma unroll
    for (int sub = 0; sub < 2; ++sub) {
      const int nT = (wave * 2 + sub) * 16;
      const float* brow = w2 + (size_t)(nT + r16) * DFEAT;
      v8f c = {};
#pragma unroll 8
      for (int kk = 0; kk < DFEAT; kk += 4) {
        v2f a = *(const v2f*)(arow + kk + klo);
        v2f bb = *(const v2f*)(brow + kk + klo);
        c = wmma4(a, bb, c);
      }
#pragma unroll
      for (int r = 0; r < 8; ++r)
        qp_lds[(rbase + r) * QP_S + nT + r16] = c[r];
    }
  }
  __syncthreads();

  // --- Phase C: scores s[q][t] = sum_d v[d]*tanh(qp[q][d]+kp[t][d]) -------
  // 16 chunks of 16 history rows; double-buffered async DMA overlaps the
  // transcendental-bound compute with the next chunk's global->LDS copy.
  const int q  = tid >> 4;
  const int tl = tid & 15;
  for (int tc = 0; tc < 16; ++tc) {
    const int buf = tc & 1;
    if (tc < 15) {
      issue_chunk(tc + 1, buf ^ 1);
      wait_asynccnt(4);      // current chunk's 4 loads done (in-order), next in flight
    } else {
      wait_asynccnt(0);
    }
    __syncthreads();

    const float4* qr = (const float4*)&qp_lds[q * QP_S];
    const float4* kr = (const float4*)&kp_lds[buf][tl * QP_S];
    const float4* vr = (const float4*)&v_lds[0];
    float acc = 0.0f;
#pragma unroll 8
    for (int d4 = 0; d4 < DFEAT / 4; ++d4) {
      float4 qv = qr[d4];
      float4 kv = kr[d4];
      float4 vv = vr[d4];
      acc += vv.x * fast_tanh(qv.x + kv.x);
      acc += vv.y * fast_tanh(qv.y + kv.y);
      acc += vv.z * fast_tanh(qv.z + kv.z);
      acc += vv.w * fast_tanh(qv.w + kv.w);
    }
    sc_lds[q * SC_S + tc * 16 + tl] = acc;
    __syncthreads();   // buf becomes writable for chunk tc+2's issue
  }

  // --- Phase D: softmax over t (wave-parallel, 2 rows per wave) -----------
  for (int qi = wave * 2; qi < wave * 2 + 2; ++qi) {
    float vals[8];
    float m = -__builtin_inff();
#pragma unroll
    for (int j = 0; j < 8; ++j) {
      vals[j] = sc_lds[qi * SC_S + lane + j * 32];
      m = fmaxf(m, vals[j]);
    }
#pragma unroll
    for (int mask = 16; mask >= 1; mask >>= 1)
      m = fmaxf(m, __shfl_xor(m, mask, 32));
    float s = 0.0f;
#pragma unroll
    for (int j = 0; j < 8; ++j) {
      float e = fast_exp2((vals[j] - m) * 1.4426950408889634f);
      vals[j] = e;
      s += e;
    }
#pragma unroll
    for (int mask = 16; mask >= 1; mask >>= 1)
      s += __shfl_xor(s, mask, 32);
    float inv = fast_rcp(s);
#pragma unroll
    for (int j = 0; j < 8; ++j)
      sc_lds[qi * SC_S + lane + j * 32] = vals[j] * inv;
  }
  __syncthreads();

  // --- Phase E: context[q][i] = sum_t alpha[q][t] * keys[b][t][i] (WMMA) --
  {
    const float* Kb = keys + (size_t)b * TH * DFEAT;
    float* orow = out + (size_t)(b * TQ + qtile * 16) * DFEAT;
#pragma unroll
    for (int sub = 0; sub < 2; ++sub) {
      const int nT = (wave * 2 + sub) * 16;
      v8f c = {};
#pragma unroll 8
      for (int kk = 0; kk < TH; kk += 4) {
        v2f a = *(const v2f*)&sc_lds[r16 * SC_S + kk + klo];  // alpha row
        v2f bb;
        bb.x = Kb[(size_t)(kk + klo) * DFEAT + nT + r16];
        bb.y = Kb[(size_t)(kk + klo + 1) * DFEAT + nT + r16];
        c = wmma4(a, bb, c);
      }
#pragma unroll
      for (int r = 0; r < 8; ++r)
        orow[(size_t)(rbase + r) * DFEAT + nT + r16] = c[r];
    }
  }
}

// ---------------------------------------------------------------------------
extern "C" void kernel_launch(void* const* d_in, const int* in_sizes, int n_in,
                              void* d_out, int out_size, void* d_ws, size_t ws_size,
                              hipStream_t stream) {
  const float* inputs = (const float*)d_in[0];  // (32,256,256)
  const float* state  = (const float*)d_in[1];  // (32,64,256)
  const float* w1     = (const float*)d_in[2];  // (256,256)
  const float* w2     = (const float*)d_in[3];  // (256,256)
  const float* v      = (const float*)d_in[4];  // (256,)
  float* out = (float*)d_out;                   // (2048,256)
  float* kproj = (float*)d_ws;                  // 8192*256*4 = 8 MB scratch

  // k_proj: 8192 tiles of 16x16, 8 per block
  kproj_gemm<<<1024, 256, 0, stream>>>(inputs, w1, kproj);
  // fused attention: one block per (batch, 16-query tile)
  fused_attn<<<NB * (TQ / 16), 256, 0, stream>>>(inputs, state, w2, v, kproj, out);
}